// Routing_16234976379165
// MI455X (gfx1250) — compile-verified
//
#include <hip/hip_runtime.h>
#include <stdint.h>

typedef __attribute__((ext_vector_type(2))) float v2f;
typedef __attribute__((ext_vector_type(8))) float v8f;
typedef __attribute__((ext_vector_type(4))) unsigned int v4u;
typedef __attribute__((ext_vector_type(8))) int  v8i;
typedef __attribute__((ext_vector_type(4))) int  v4i;

#define CAPS   1152
#define LL     16
#define ELEMS  (CAPS * LL)   /* 18432 floats = 73728 bytes per (n,c) problem */
#define TPB    256
#define NWAVES 8
#define TILES_PER_WAVE (CAPS / 16 / NWAVES)  /* 72 tiles / 8 waves = 9 */

// dynamic LDS layout (floats):
//   Xs   : ELEMS        (72 KB)   problem tile, row-major [cap][l]
//   bArr : CAPS         logits -> exp -> c (in place)
//   red  : TPB          reduction scratch
//   sjv  : 16
//   vjv  : 16
#define SMEM_FLOATS (ELEMS + CAPS + TPB + 16 + 16)

__global__ __launch_bounds__(TPB) void routing_caps_kernel(
    const float* __restrict__ X, float* __restrict__ out) {
  extern __shared__ float smem[];
  float* Xs   = smem;
  float* bArr = Xs + ELEMS;
  float* red  = bArr + CAPS;
  float* sjv  = red + TPB;
  float* vjv  = sjv + 16;

  const int tid  = threadIdx.x;
  const int p    = blockIdx.x;          // 0..2047  (n*C + c)
  const int lane = tid & 31;
  const int wave = tid >> 5;

  // ---------------- Phase 0: TDM async load of full problem tile ----------
  const float* Xp = X + (size_t)p * ELEMS;
  if (tid == 0) {
    uint64_t ga = (uint64_t)(uintptr_t)Xp;
    uint32_t ldsa = (uint32_t)(uintptr_t)Xs;   // LDS byte offset (addr[31:0])
    // D# group 0: count=1 | lds_addr | global_addr[56:0] | type=2
    v4u g0 = { 1u,
               ldsa,
               (uint32_t)ga,
               (uint32_t)((ga >> 32) & 0x1FFFFFFu) | 0x80000000u };
    // D# group 1: data_size=4B; 1-D: tensor_dim0=tile_dim0=ELEMS, dim1=1
    v8i g1 = { (int)(2u << 16),                          // data_size code 2 (4B)
               (int)(((uint32_t)ELEMS & 0xFFFFu) << 16), // tensor_dim0[15:0]
               (int)(1u << 16),                          // dim0 hi=0 | tensor_dim1=1
               (int)(((uint32_t)ELEMS & 0xFFFFu) << 16), // dim1 hi=0 | tile_dim0
               0,                                        // tile_dim1=0, tile_dim2=0
               (int)(uint32_t)ELEMS,                     // tensor_dim0_stride lo32
               0,                                        // stride hi | dim1_stride lo
               0 };                                      // dim1_stride hi
    v4i z4 = { 0, 0, 0, 0 };
    v8i z8 = { 0, 0, 0, 0, 0, 0, 0, 0 };
    __builtin_amdgcn_tensor_load_to_lds(g0, g1, z4, z4, z8, 0);
  }
  __builtin_amdgcn_s_wait_tensorcnt(0);   // issuing wave waits; others: cnt==0
  __syncthreads();

  // ---------------- Iteration 1: b == 0 -> c uniform = 1/1152 -------------
  {
    const int col = tid & 15;
    const int g   = tid >> 4;            // 16 row groups
    float acc = 0.0f;
    for (int r = g; r < CAPS; r += 16) acc += Xs[r * LL + col];
    red[tid] = acc;
    __syncthreads();
    if (tid < 16) {
      float s = 0.0f;
      #pragma unroll
      for (int g2 = 0; g2 < 16; ++g2) s += red[g2 * 16 + tid];
      sjv[tid] = s * (1.0f / (float)CAPS);
    }
    __syncthreads();
    if (tid == 0) {
      float sq = 0.0f;
      #pragma unroll
      for (int l = 0; l < LL; ++l) sq += sjv[l] * sjv[l];
      float scale = sq / ((sq + 1.0f) * (sq + 1e-8f));   // squash (faithful)
      #pragma unroll
      for (int l = 0; l < LL; ++l) vjv[l] = sjv[l] * scale;
    }
    __syncthreads();
  }

  // --------- bj = X . vj  via v_wmma_f32_16x16x4_f32 (16 caps / tile) -----
  {
    const int m  = lane & 15;
    const int hi = lane >> 4;            // 0: K={0,1}, 1: K={2,3} per chunk
    for (int t = wave * TILES_PER_WAVE; t < (wave + 1) * TILES_PER_WAVE; ++t) {
      const int base = t * 16;
      v8f d = (v8f)0.0f;
      #pragma unroll
      for (int kc = 0; kc < 4; ++kc) {
        const int k0 = kc * 4 + hi * 2;
        v2f a = *(const v2f*)&Xs[(base + m) * LL + k0];  // A[m][k0..k0+1]
        v2f b;
        b.x = vjv[k0];                                   // B rows = vj chunk,
        b.y = vjv[k0 + 1];                               // identical columns
        d = __builtin_amdgcn_wmma_f32_16x16x4_f32(
                false, a, false, b, (short)0, d, false, false);
      }
      // D layout: lane 0 col0 holds rows 0..7 (vgpr 0..7), lane 16 rows 8..15
      if (lane == 0) {
        #pragma unroll
        for (int r = 0; r < 8; ++r) bArr[base + r] = d[r];
      } else if (lane == 16) {
        #pragma unroll
        for (int r = 0; r < 8; ++r) bArr[base + 8 + r] = d[r];
      }
    }
    __syncthreads();
  }

  // ---------------- Iteration 2: softmax(b) -> sj -> vj (output) ----------
  {
    float mx = -3.402823466e38f;
    for (int i = tid; i < CAPS; i += TPB) mx = fmaxf(mx, bArr[i]);
    red[tid] = mx;
    __syncthreads();
    for (int s = TPB / 2; s > 0; s >>= 1) {
      if (tid < s) red[tid] = fmaxf(red[tid], red[tid + s]);
      __syncthreads();
    }
    const float gmax = red[0];
    float se = 0.0f;
    for (int i = tid; i < CAPS; i += TPB) {
      float e = __expf(bArr[i] - gmax);
      bArr[i] = e;
      se += e;
    }
    __syncthreads();          // all reads of red[0] done before reuse
    red[tid] = se;
    __syncthreads();
    for (int s = TPB / 2; s > 0; s >>= 1) {
      if (tid < s) red[tid] += red[tid + s];
      __syncthreads();
    }
    const float inv = 1.0f / red[0];
    for (int i = tid; i < CAPS; i += TPB) bArr[i] *= inv;   // c coefficients
    __syncthreads();

    const int col = tid & 15;
    const int g   = tid >> 4;
    float acc = 0.0f;
    for (int r = g; r < CAPS; r += 16) acc += bArr[r] * Xs[r * LL + col];
    red[tid] = acc;
    __syncthreads();
    if (tid < 16) {
      float s = 0.0f;
      #pragma unroll
      for (int g2 = 0; g2 < 16; ++g2) s += red[g2 * 16 + tid];
      sjv[tid] = s;
    }
    __syncthreads();
    if (tid == 0) {
      float sq = 0.0f;
      #pragma unroll
      for (int l = 0; l < LL; ++l) sq += sjv[l] * sjv[l];
      float scale = sq / ((sq + 1.0f) * (sq + 1e-8f));
      #pragma unroll
      for (int l = 0; l < LL; ++l) vjv[l] = sjv[l] * scale;
    }
    __syncthreads();
    if (tid < LL) out[(size_t)p * LL + tid] = vjv[tid];
  }
}

extern "C" void kernel_launch(void* const* d_in, const int* in_sizes, int n_in,
                              void* d_out, int out_size, void* d_ws, size_t ws_size,
                              hipStream_t stream) {
  (void)n_in; (void)out_size; (void)d_ws; (void)ws_size;
  const float* X = (const float*)d_in[0];
  float* out = (float*)d_out;
  const int nproblems = in_sizes[0] / ELEMS;   // 64*32 = 2048
  const size_t shmem = (size_t)SMEM_FLOATS * sizeof(float);  // ~79.5 KB
  (void)hipFuncSetAttribute(reinterpret_cast<const void*>(routing_caps_kernel),
                            hipFuncAttributeMaxDynamicSharedMemorySize, (int)shmem);
  routing_caps_kernel<<<nproblems, TPB, shmem, stream>>>(X, out);
}